// SpatialEmbedding_40303973105896
// MI455X (gfx1250) — compile-verified
//
#include <hip/hip_runtime.h>

// ---------------------------------------------------------------------------
// SpatialEmbedding fused kernel for MI455X (gfx1250, wave32)
//   phase 1: ball-query (first-K by index, radius 5/480) on per-batch xy in LDS
//   phase 2: delta-embedding -> MLP (32->128 relu ->64) via V_WMMA_F32_16X16X4_F32
// ---------------------------------------------------------------------------

#define BATCH          8
#define NPTS           4096
#define KNN            16
#define INC            32      // KNN * 2
#define HID            128
#define OUTC           64
#define ROWS_PER_BLOCK 128
#define THREADS        128     // 4 waves of 32
#define EMB_STRIDE     33      // odd -> bank-conflict-free A-frag reads
#define H_STRIDE       129     // odd -> bank-conflict-free A-frag reads

typedef float v2f __attribute__((ext_vector_type(2)));
typedef float v8f __attribute__((ext_vector_type(8)));

__global__ __launch_bounds__(THREADS)
void spatial_embed_kernel(const float* __restrict__ xytp,
                          const float* __restrict__ W1,
                          const float* __restrict__ b1,
                          const float* __restrict__ W2,
                          const float* __restrict__ b2,
                          float* __restrict__ out)
{
    const float R2 = (5.0f / 480.0f) * (5.0f / 480.0f);

    extern __shared__ float smem[];
    float* xy_s  = smem;                                 // 4096*2      = 8192 f
    float* emb_s = xy_s  + NPTS * 2;                     // 128*33      = 4224 f
    float* W1_s  = emb_s + ROWS_PER_BLOCK * EMB_STRIDE;  // 32*128      = 4096 f
    float* W2_s  = W1_s  + INC * HID;                    // 128*64      = 8192 f
    float* b1_s  = W2_s  + HID * OUTC;                   // 128 f
    float* b2_s  = b1_s  + HID;                          // 64 f
    float* hbuf  = b2_s  + OUTC;                         // 4*16*129    = 8256 f

    const int tid   = threadIdx.x;
    const int batch = blockIdx.y;
    const int rbase = blockIdx.x * ROWS_PER_BLOCK;

    // ---- cooperative loads into LDS -------------------------------------
    const float4* xy4 = (const float4*)xytp + (size_t)batch * NPTS;
    for (int m = tid; m < NPTS; m += THREADS) {
        float4 v = xy4[m];          // xytp[b][m] = (x, y_used, t_used, p)
        xy_s[2 * m]     = v.y;      // channel 1
        xy_s[2 * m + 1] = v.z;      // channel 2
    }
    for (int i = tid; i < INC * HID; i += THREADS)  W1_s[i] = W1[i];
    for (int i = tid; i < HID * OUTC; i += THREADS) W2_s[i] = W2[i];
    if (tid < HID)  b1_s[tid] = b1[tid];
    if (tid < OUTC) b2_s[tid] = b2[tid];
    __syncthreads();

    // ---- ball query: first-K neighbors in ascending index order ---------
    {
        const int n   = rbase + tid;        // this thread's query point
        const float qx = xy_s[2 * n];
        const float qy = xy_s[2 * n + 1];
        float* erow = emb_s + tid * EMB_STRIDE;
        int cnt = 0;
        for (int m = 0; m < NPTS; ++m) {    // broadcast LDS read per iter
            float dx = qx - xy_s[2 * m];
            float dy = qy - xy_s[2 * m + 1];
            float d2 = dx * dx + dy * dy;
            if (d2 < R2 && cnt < KNN) {     // self (m==n) included: d2==0
                erow[2 * cnt]     = dx;
                erow[2 * cnt + 1] = dy;
                ++cnt;
            }
        }
        for (; cnt < KNN; ++cnt) {          // empty slots -> self -> delta 0
            erow[2 * cnt]     = 0.0f;
            erow[2 * cnt + 1] = 0.0f;
        }
    }
    __syncthreads();

    // ---- MLP via fp32 WMMA (16x16x4) ------------------------------------
    const int lane = tid & 31;
    const int wave = tid >> 5;
    const int lrow = lane & 15;    // M index within 16-row tile
    const int half = lane >> 4;    // half-wave selects K pair {0,1} vs {2,3}
    float* hw = hbuf + wave * (16 * H_STRIDE);

    for (int t = 0; t < 2; ++t) {                 // 2 row tiles per wave
        const int tilerow = wave * 32 + t * 16;   // local row base

        // layer 1: emb(16x32) @ W1(32x128) + b1, relu -> hw(16x128)
        for (int j = 0; j < HID / 16; ++j) {
            v8f acc;
            {
                float bv = b1_s[j * 16 + lrow];
#pragma unroll
                for (int r = 0; r < 8; ++r) acc[r] = bv;
            }
            const float* arow = emb_s + (tilerow + lrow) * EMB_STRIDE + half * 2;
            const float* bcol = W1_s + (half * 2) * HID + j * 16 + lrow;
#pragma unroll
            for (int k0 = 0; k0 < INC / 4; ++k0) {
                v2f a, bb;
                a.x  = arow[k0 * 4];
                a.y  = arow[k0 * 4 + 1];
                bb.x = bcol[(k0 * 4) * HID];
                bb.y = bcol[(k0 * 4) * HID + HID];
                acc = __builtin_amdgcn_wmma_f32_16x16x4_f32(
                        false, a, false, bb, (short)0, acc, false, false);
            }
#pragma unroll
            for (int r = 0; r < 8; ++r) {
                float v = acc[r] > 0.0f ? acc[r] : 0.0f;   // relu
                hw[(r + 8 * half) * H_STRIDE + j * 16 + lrow] = v;
            }
        }
        __syncthreads();   // uniform: all waves run same trip counts

        // layer 2: hw(16x128) @ W2(128x64) + b2 -> out
        for (int j = 0; j < OUTC / 16; ++j) {
            v8f acc;
            {
                float bv = b2_s[j * 16 + lrow];
#pragma unroll
                for (int r = 0; r < 8; ++r) acc[r] = bv;
            }
            const float* arow = hw + lrow * H_STRIDE + half * 2;
            const float* bcol = W2_s + (half * 2) * OUTC + j * 16 + lrow;
#pragma unroll
            for (int k0 = 0; k0 < HID / 4; ++k0) {
                v2f a, bb;
                a.x  = arow[k0 * 4];
                a.y  = arow[k0 * 4 + 1];
                bb.x = bcol[(k0 * 4) * OUTC];
                bb.y = bcol[(k0 * 4) * OUTC + OUTC];
                acc = __builtin_amdgcn_wmma_f32_16x16x4_f32(
                        false, a, false, bb, (short)0, acc, false, false);
            }
#pragma unroll
            for (int r = 0; r < 8; ++r) {
                const int rg = rbase + tilerow + r + 8 * half;
                out[((size_t)batch * NPTS + rg) * OUTC + j * 16 + lrow] = acc[r];
            }
        }
        __syncthreads();   // before next tile reuses hw
    }
}

extern "C" void kernel_launch(void* const* d_in, const int* in_sizes, int n_in,
                              void* d_out, int out_size, void* d_ws, size_t ws_size,
                              hipStream_t stream) {
    const float* xytp = (const float*)d_in[0];
    const float* W1   = (const float*)d_in[1];
    const float* b1   = (const float*)d_in[2];
    const float* W2   = (const float*)d_in[3];
    const float* b2   = (const float*)d_in[4];
    float* out = (float*)d_out;

    const size_t smem = (size_t)(NPTS * 2 +
                                 ROWS_PER_BLOCK * EMB_STRIDE +
                                 INC * HID +
                                 HID * OUTC +
                                 HID + OUTC +
                                 4 * 16 * H_STRIDE) * sizeof(float);

    dim3 grid(NPTS / ROWS_PER_BLOCK, BATCH);
    dim3 block(THREADS);
    hipLaunchKernelGGL(spatial_embed_kernel, grid, block, smem, stream,
                       xytp, W1, b1, W2, b2, out);
}